// SVDPPEmbedding_67688684585005
// MI455X (gfx1250) — compile-verified
//
#include <hip/hip_runtime.h>

typedef float v2f __attribute__((ext_vector_type(2)));
typedef float v4f __attribute__((ext_vector_type(4)));
typedef float v8f __attribute__((ext_vector_type(8)));

#define WPB 4            // waves per block
#define EPW 16           // batch elements per wave == WMMA tile M/N
#define LSTRIDE 68       // LDS row stride (floats): conflict-free row/col access
#define UF 8             // gather unroll: 8 B128 loads in flight per wave

__global__ __launch_bounds__(WPB * 32)
void svdpp_fused_kernel(const int* __restrict__ user_ids,
                        const int* __restrict__ item_ids,
                        const int* __restrict__ offsets,
                        const int* __restrict__ flat_implicit,
                        const float* __restrict__ user_emb,
                        const float* __restrict__ item_emb,
                        const float* __restrict__ imp_emb,
                        const float* __restrict__ user_bias,
                        const float* __restrict__ item_bias,
                        const float* __restrict__ global_bias,
                        float* __restrict__ out,
                        int B, int N)
{
    __shared__ float sFU[WPB][EPW][LSTRIDE];
    __shared__ float sIT[WPB][EPW][LSTRIDE];

    const int lane = threadIdx.x & 31;
    const int wave = threadIdx.x >> 5;
    const int base = (blockIdx.x * WPB + wave) * EPW;

    const int half = lane >> 4;        // which of the 2 rows per loop slot
    const int c4   = (lane & 15) * 4;  // float4 chunk within the 64-dim row

    float (*FU)[LSTRIDE] = sFU[wave];
    float (*IT)[LSTRIDE] = sIT[wave];

    // ---------------- Phase A: ragged gather + segment sum ----------------
    for (int t = 0; t < EPW; ++t) {
        const int b = base + t;
        if (b >= B) break;                       // wave-uniform
        const int start = offsets[b];
        const int end   = (b + 1 < B) ? offsets[b + 1] : N;
        const int size  = end - start;

        v4f acc0 = {0.f,0.f,0.f,0.f};
        v4f acc1 = acc0, acc2 = acc0, acc3 = acc0;

        for (int j = start; j < end; j += 2 * UF) {
            int  idx[UF];
            bool ok[UF];
            v4f  v[UF];
#pragma unroll
            for (int u = 0; u < UF; ++u) {
                int r  = j + 2 * u + half;
                ok[u]  = (r < end);
                int rs = ok[u] ? r : start;       // clamp: keep load unconditional
                idx[u] = flat_implicit[rs];
            }
#pragma unroll
            for (int u = 0; u < UF; ++u)          // 8 independent B128 gathers
                v[u] = *(const v4f*)(imp_emb + (size_t)idx[u] * 64 + c4);
#pragma unroll
            for (int u = 0; u < UF; ++u) {
                v4f z = {0.f,0.f,0.f,0.f};
                v4f m = ok[u] ? v[u] : z;         // select-mask, no divergence
                if ((u & 3) == 0) acc0 += m;
                else if ((u & 3) == 1) acc1 += m;
                else if ((u & 3) == 2) acc2 += m;
                else acc3 += m;
            }
        }
        v4f acc = (acc0 + acc1) + (acc2 + acc3);
        // lanes l and l^16 accumulated disjoint rows of the same d-chunk
        acc.x += __shfl_xor(acc.x, 16, 32);
        acc.y += __shfl_xor(acc.y, 16, 32);
        acc.z += __shfl_xor(acc.z, 16, 32);
        acc.w += __shfl_xor(acc.w, 16, 32);

        if (half == 0) {
            const float inv = (size > 0) ? rsqrtf((float)size) : 1.0f;
            const int uid = user_ids[b];
            const int iid = item_ids[b];
            v4f u4 = *(const v4f*)(user_emb + (size_t)uid * 64 + c4);
            v4f i4 = *(const v4f*)(item_emb + (size_t)iid * 64 + c4);
            v4f fu = u4 + acc * inv;
            *(v4f*)&FU[t][c4] = fu;               // final_user row t
            *(v4f*)&IT[t][c4] = i4;               // item row t
        }
    }

    __syncthreads();

    // ---------------- Phase B: dots on the matrix pipe ----------------
    // D(16x16) = FU(16x64) x IT^T(64x16), 16 K=4 steps of v_wmma_f32_16x16x4_f32.
    // diag(D)[t] = dot(final_user[t], item[t]).
    // A layout: lane l holds A[l&15][k0 + (l>=16 ? 2 : 0) + {0,1}] in 2 VGPRs.
    // B layout: lane l holds B[same k][n = l&15] = IT[l&15][same k].
    const int m  = lane & 15;
    const int hk = half * 2;
    v8f d = {0.f,0.f,0.f,0.f,0.f,0.f,0.f,0.f};
#pragma unroll
    for (int kk = 0; kk < 16; ++kk) {
        const int kb = kk * 4 + hk;
        v2f a, bb;
        a.x  = FU[m][kb];
        a.y  = FU[m][kb + 1];
        bb.x = IT[m][kb];
        bb.y = IT[m][kb + 1];
        d = __builtin_amdgcn_wmma_f32_16x16x4_f32(false, a, false, bb,
                                                  (short)0, d, false, false);
    }

    // Diagonal: t<8 -> VGPR t, lane t ; t>=8 -> VGPR t-8, lane t+16.
    const bool owner = (lane < 8) || (lane >= 24);
    const int  t     = (lane < 8) ? lane : (lane - 16);
    const int  b     = base + t;
    if (owner && b < B) {
        const int comp = lane & 7;
        float dot = d[0];
        dot = (comp == 1) ? d[1] : dot;
        dot = (comp == 2) ? d[2] : dot;
        dot = (comp == 3) ? d[3] : dot;
        dot = (comp == 4) ? d[4] : dot;
        dot = (comp == 5) ? d[5] : dot;
        dot = (comp == 6) ? d[6] : dot;
        dot = (comp == 7) ? d[7] : dot;
        out[b] = dot + user_bias[user_ids[b]] + item_bias[item_ids[b]]
                     + global_bias[0];
    }
}

extern "C" void kernel_launch(void* const* d_in, const int* in_sizes, int n_in,
                              void* d_out, int out_size, void* d_ws, size_t ws_size,
                              hipStream_t stream) {
    (void)n_in; (void)out_size; (void)d_ws; (void)ws_size;
    const int*   user_ids      = (const int*)d_in[0];
    const int*   item_ids      = (const int*)d_in[1];
    const int*   offsets       = (const int*)d_in[2];
    const int*   flat_implicit = (const int*)d_in[3];
    const float* user_emb      = (const float*)d_in[4];
    const float* item_emb      = (const float*)d_in[5];
    const float* imp_emb       = (const float*)d_in[6];
    const float* user_bias     = (const float*)d_in[7];
    const float* item_bias     = (const float*)d_in[8];
    const float* global_bias   = (const float*)d_in[9];
    float* out = (float*)d_out;

    const int B = in_sizes[0];
    const int N = in_sizes[3];
    const int elems_per_block = WPB * EPW;          // 64
    const int grid = (B + elems_per_block - 1) / elems_per_block;

    svdpp_fused_kernel<<<dim3(grid), dim3(WPB * 32), 0, stream>>>(
        user_ids, item_ids, offsets, flat_implicit,
        user_emb, item_emb, imp_emb,
        user_bias, item_bias, global_bias,
        out, B, N);
}